// MultiScaleGNN_90271622627427
// MI455X (gfx1250) — compile-verified
//
#include <hip/hip_runtime.h>
#include <math.h>

// ---------------------------------------------------------------------------
// MultiScaleGNN (3x GAT + JK-BiLSTM attention) for MI455X / gfx1250.
//
// Design notes (MI455X, wave32, WMMA):
//  * Dense GEMMs use V_WMMA_F32_16X16X4_F32: fp32-exact, 16x16 C tile per
//    wave, K stepped by 4. N=50000 nodes = 3125 exact 16-row tiles.
//  * Edge softmax/aggregation is the bandwidth-dominant part (~1.3 GB of
//    gather+atomic traffic total): wave-per-edge, coalesced 64-float feature
//    rows, GLOBAL_ATOMIC_ADD_F32 scatter, order-preserving-uint atomic max.
//  * All scratch lives in d_ws; all launches on `stream` (graph-capturable).
// ---------------------------------------------------------------------------

#define NEG_SLOPE 0.2f
#define HID 64
#define LSTMH 96
#define G4 384   // 4 * LSTMH gate columns

typedef __attribute__((ext_vector_type(2))) float v2f;
typedef __attribute__((ext_vector_type(8))) float v8f;

// ---- order-preserving float<->uint for atomic max over signed floats ------
__device__ __forceinline__ unsigned f2ord(float f) {
  unsigned b = __float_as_uint(f);
  return (b & 0x80000000u) ? ~b : (b | 0x80000000u);
}
__device__ __forceinline__ float ord2f(unsigned u) {
  return __uint_as_float((u & 0x80000000u) ? (u ^ 0x80000000u) : ~u);
}

// ---------------------------------------------------------------------------
// Generic fp32 WMMA GEMM:  C[M,Nc] (=|+=) A[M,K] @ W[Nc,K]^T (+ b0 + b1), relu
// One wave per 16x16 C tile.  Fragment layouts per CDNA5 ISA 7.12.2:
//   A 16x4 : lanes 0-15 row M=lane   (v0=K0,v1=K1); lanes 16-31 (v0=K2,v1=K3)
//   B 4x16 : lanes 0-15 col N=lane   (v0=K0,v1=K1); lanes 16-31 (v0=K2,v1=K3)
//   C/D    : VGPR i -> M = i + 8*(lane>=16), N = lane&15
// K must be a multiple of 4 (128/64/96/384 all are), M a multiple of 16.
// ---------------------------------------------------------------------------
__global__ void wmma_gemm_k(const float* __restrict__ A,
                            const float* __restrict__ W,
                            const float* __restrict__ b0,
                            const float* __restrict__ b1,
                            float* __restrict__ C,
                            int M, int K, int Nc, int accumulate, int relu) {
  const int lane = threadIdx.x;                       // 0..31 (wave32)
  const int rt   = blockIdx.x * blockDim.y + threadIdx.y;
  const int ct   = blockIdx.y;
  const int rowBase = rt * 16;
  const int colBase = ct * 16;
  if (rowBase >= M) return;
  const int hi  = lane >> 4;
  const int l16 = lane & 15;
  const int row = rowBase + l16;
  const int col = colBase + l16;

  // 8-byte aligned: row*K and col*K are multiples of 4 floats, +hi*2 even.
  const float2* __restrict__ ap =
      reinterpret_cast<const float2*>(A + (size_t)row * K + hi * 2);
  const float2* __restrict__ wp =
      reinterpret_cast<const float2*>(W + (size_t)col * K + hi * 2);

  v8f acc;
  if (accumulate) {
#pragma unroll
    for (int i = 0; i < 8; ++i)
      acc[i] = C[(size_t)(rowBase + i + hi * 8) * Nc + col];
  } else {
    float bb = 0.f;
    if (b0) bb += b0[col];
    if (b1) bb += b1[col];
#pragma unroll
    for (int i = 0; i < 8; ++i) acc[i] = bb;
  }

  const int kq = K >> 2;
#pragma unroll 4
  for (int k = 0; k < kq; ++k) {
    float2 av = ap[2 * k];          // A[row][k*4 + 2*hi + {0,1}]
    float2 wv = wp[2 * k];          // W[col][k*4 + 2*hi + {0,1}]
    v2f a; a[0] = av.x; a[1] = av.y;
    v2f b; b[0] = wv.x; b[1] = wv.y;
    acc = __builtin_amdgcn_wmma_f32_16x16x4_f32(
        /*neg_a=*/false, a, /*neg_b=*/false, b,
        /*c_mod=*/(short)0, acc, /*reuse_a=*/false, /*reuse_b=*/false);
  }

#pragma unroll
  for (int i = 0; i < 8; ++i) {
    float v = acc[i];
    if (relu) v = fmaxf(v, 0.f);
    C[(size_t)(rowBase + i + hi * 8) * Nc + col] = v;
  }
}

// ---- al[n] = h[n,:].asrc ; ar[n] = h[n,:].adst  (wave per node) -----------
__global__ void alar_k(const float* __restrict__ h,
                       const float* __restrict__ asrc,
                       const float* __restrict__ adst,
                       float* __restrict__ al, float* __restrict__ ar, int N) {
  const int lane = threadIdx.x & 31;
  const int n = (int)(((size_t)blockIdx.x * blockDim.x + threadIdx.x) >> 5);
  if (n >= N) return;
  float h0 = h[(size_t)n * HID + lane];
  float h1 = h[(size_t)n * HID + 32 + lane];
  float sa = h0 * asrc[lane] + h1 * asrc[lane + 32];
  float sb = h0 * adst[lane] + h1 * adst[lane + 32];
#pragma unroll
  for (int o = 16; o > 0; o >>= 1) {
    sa += __shfl_xor(sa, o, 32);
    sb += __shfl_xor(sb, o, 32);
  }
  if (lane == 0) { al[n] = sa; ar[n] = sb; }
}

// ---- pass A: segment max of leaky_relu logits over dst --------------------
__global__ void edge_max_k(const int* __restrict__ srcl, const int* __restrict__ dstl,
                           const float* __restrict__ al, const float* __restrict__ ar,
                           unsigned* __restrict__ mu, int E, int TE) {
  int e = blockIdx.x * blockDim.x + threadIdx.x;
  if (e >= TE) return;
  int s = (e < E) ? srcl[e] : (e - E);   // self loops appended
  int d = (e < E) ? dstl[e] : (e - E);
  float z = al[s] + ar[d];
  float lr = (z > 0.f) ? z : NEG_SLOPE * z;
  atomicMax(mu + d, f2ord(lr));
}

// ---- pass B: segment sum of exp(lr - m[dst]) ------------------------------
__global__ void edge_den_k(const int* __restrict__ srcl, const int* __restrict__ dstl,
                           const float* __restrict__ al, const float* __restrict__ ar,
                           const unsigned* __restrict__ mu, float* __restrict__ den,
                           int E, int TE) {
  int e = blockIdx.x * blockDim.x + threadIdx.x;
  if (e >= TE) return;
  int s = (e < E) ? srcl[e] : (e - E);
  int d = (e < E) ? dstl[e] : (e - E);
  float z = al[s] + ar[d];
  float lr = (z > 0.f) ? z : NEG_SLOPE * z;
  atomicAdd(den + d, expf(lr - ord2f(mu[d])));
}

// ---- pass C: acc[dst,:] += coef * h[src,:]  (wave per edge, coalesced) ----
__global__ void edge_agg_k(const int* __restrict__ srcl, const int* __restrict__ dstl,
                           const float* __restrict__ al, const float* __restrict__ ar,
                           const unsigned* __restrict__ mu, const float* __restrict__ den,
                           const float* __restrict__ h, float* __restrict__ acc,
                           int E, int TE) {
  const int lane = threadIdx.x & 31;
  const int e = (int)(((size_t)blockIdx.x * blockDim.x + threadIdx.x) >> 5);
  if (e >= TE) return;
  int s = (e < E) ? srcl[e] : (e - E);
  int d = (e < E) ? dstl[e] : (e - E);
  float z = al[s] + ar[d];
  float lr = (z > 0.f) ? z : NEG_SLOPE * z;
  float coef = expf(lr - ord2f(mu[d])) / den[d];
  size_t sb = (size_t)s * HID, db = (size_t)d * HID;
  atomicAdd(acc + db + lane,      coef * h[sb + lane]);
  atomicAdd(acc + db + 32 + lane, coef * h[sb + 32 + lane]);
}

// ---- seq[l] = relu(acc + b) ----------------------------------------------
__global__ void bias_relu_k(const float* __restrict__ acc, const float* __restrict__ b,
                            float* __restrict__ out, size_t total) {
  size_t i = (size_t)blockIdx.x * blockDim.x + threadIdx.x;
  if (i >= total) return;
  out[i] = fmaxf(acc[i] + b[i & 63], 0.f);
}

// ---- LSTM elementwise gate update (i,f,g,o split of 384) ------------------
__global__ void lstm_pw_k(const float* __restrict__ g, float* __restrict__ h,
                          float* __restrict__ c, int N) {
  size_t i = (size_t)blockIdx.x * blockDim.x + threadIdx.x;
  if (i >= (size_t)N * LSTMH) return;
  int n = (int)(i / LSTMH), j = (int)(i - (size_t)n * LSTMH);
  const float* gr = g + (size_t)n * G4;
  float gi = gr[j], gf = gr[96 + j], gg = gr[192 + j], go = gr[288 + j];
  float si = 1.f / (1.f + expf(-gi));
  float sf = 1.f / (1.f + expf(-gf));
  float so = 1.f / (1.f + expf(-go));
  float cv = sf * c[i] + si * tanhf(gg);
  c[i] = cv;
  h[i] = so * tanhf(cv);
}

// ---- per-direction attention-logit contribution: h[n,:96] . aw ------------
__global__ void logit_k(const float* __restrict__ h, const float* __restrict__ aw,
                        float* __restrict__ out, int N) {
  const int lane = threadIdx.x & 31;
  const int n = (int)(((size_t)blockIdx.x * blockDim.x + threadIdx.x) >> 5);
  if (n >= N) return;
  const float* hr = h + (size_t)n * LSTMH;
  float s = hr[lane] * aw[lane] + hr[32 + lane] * aw[32 + lane] +
            hr[64 + lane] * aw[64 + lane];
#pragma unroll
  for (int o = 16; o > 0; o >>= 1) s += __shfl_xor(s, o, 32);
  if (lane == 0) out[n] = s;
}

// ---- JK attention: softmax over L=3, agg = sum_t alpha_t * seq[t] ---------
__global__ void attn_k(const float* __restrict__ lf, const float* __restrict__ lb,
                       const float* __restrict__ attb, const float* __restrict__ seq,
                       float* __restrict__ agg, int N) {
  const int lane = threadIdx.x & 31;
  const int n = (int)(((size_t)blockIdx.x * blockDim.x + threadIdx.x) >> 5);
  if (n >= N) return;
  float ab = attb[0];
  float l0 = lf[n] + lb[n] + ab;
  float l1 = lf[(size_t)N + n] + lb[(size_t)N + n] + ab;
  float l2 = lf[2 * (size_t)N + n] + lb[2 * (size_t)N + n] + ab;
  float mx = fmaxf(l0, fmaxf(l1, l2));
  float e0 = expf(l0 - mx), e1 = expf(l1 - mx), e2 = expf(l2 - mx);
  float inv = 1.f / (e0 + e1 + e2);
  float a0 = e0 * inv, a1 = e1 * inv, a2 = e2 * inv;
  size_t base = (size_t)n * HID;
  size_t LN = (size_t)N * HID;
  agg[base + lane] = a0 * seq[base + lane] + a1 * seq[LN + base + lane] +
                     a2 * seq[2 * LN + base + lane];
  agg[base + 32 + lane] = a0 * seq[base + 32 + lane] +
                          a1 * seq[LN + base + 32 + lane] +
                          a2 * seq[2 * LN + base + 32 + lane];
}

__global__ void fill_k(float* __restrict__ p, float v, size_t n) {
  size_t i = (size_t)blockIdx.x * blockDim.x + threadIdx.x;
  if (i < n) p[i] = v;
}

// ---------------------------------------------------------------------------
extern "C" void kernel_launch(void* const* d_in, const int* in_sizes, int n_in,
                              void* d_out, int out_size, void* d_ws, size_t ws_size,
                              hipStream_t stream) {
  (void)n_in; (void)out_size; (void)ws_size;
  const float* x  = (const float*)d_in[0];
  const int*   ei = (const int*)d_in[1];
  const int N  = in_sizes[0] / 128;   // 50000 (divisible by 16)
  const int E  = in_sizes[1] / 2;     // 800000
  const int TE = E + N;               // + self loops
  const int* srcl = ei;
  const int* dstl = ei + E;

  // ---- workspace layout (floats) ----
  float* ws = (float*)d_ws;
  size_t off = 0;
  float*    seq   = ws + off; off += (size_t)3 * N * HID;   // layer outputs
  float*    hbuf  = ws + off; off += (size_t)N * HID;       // GAT transformed h
  float*    accum = ws + off; off += (size_t)N * HID;       // aggregation / agg
  float*    al    = ws + off; off += (size_t)N;
  float*    ar    = ws + off; off += (size_t)N;
  unsigned* mu    = (unsigned*)(ws + off); off += (size_t)N;
  float*    den   = ws + off; off += (size_t)N;
  float*    gates = ws + off; off += (size_t)N * G4;
  float*    lh    = ws + off; off += (size_t)N * LSTMH;
  float*    lc    = ws + off; off += (size_t)N * LSTMH;
  float*    lf    = ws + off; off += (size_t)3 * N;
  float*    lb    = ws + off; off += (size_t)3 * N;

  const dim3 gb(32, 4, 1);  // 4 waves per block, one 16x16 tile each
  auto gemm = [&](const float* A, const float* W, const float* b0,
                  const float* b1, float* C, int M, int K, int Nc,
                  int acc, int relu) {
    int Mt = (M + 15) / 16;
    dim3 grid((Mt + 3) / 4, Nc / 16, 1);
    wmma_gemm_k<<<grid, gb, 0, stream>>>(A, W, b0, b1, C, M, K, Nc, acc, relu);
  };
  auto fill = [&](float* p, float v, size_t n) {
    fill_k<<<(unsigned)((n + 255) / 256), 256, 0, stream>>>(p, v, n);
  };

  // ================= GAT layers =================
  const float* xin = x;
  int K = 128;
  for (int l = 0; l < 3; ++l) {
    const float* W  = (const float*)d_in[2 + 4 * l];
    const float* as = (const float*)d_in[3 + 4 * l];
    const float* ad = (const float*)d_in[4 + 4 * l];
    const float* bc = (const float*)d_in[5 + 4 * l];

    gemm(xin, W, nullptr, nullptr, hbuf, N, K, HID, 0, 0);   // h = x @ W^T
    alar_k<<<(unsigned)(((size_t)N * 32 + 255) / 256), 256, 0, stream>>>(
        hbuf, as, ad, al, ar, N);
    fill((float*)mu, 0.0f, N);            // ord-uint 0 == most negative float
    fill(den, 0.0f, N);
    fill(accum, 0.0f, (size_t)N * HID);
    edge_max_k<<<(TE + 255) / 256, 256, 0, stream>>>(srcl, dstl, al, ar, mu, E, TE);
    edge_den_k<<<(TE + 255) / 256, 256, 0, stream>>>(srcl, dstl, al, ar, mu, den, E, TE);
    edge_agg_k<<<(unsigned)(((size_t)TE * 32 + 255) / 256), 256, 0, stream>>>(
        srcl, dstl, al, ar, mu, den, hbuf, accum, E, TE);
    float* sl = seq + (size_t)l * N * HID;
    bias_relu_k<<<(unsigned)(((size_t)N * HID + 255) / 256), 256, 0, stream>>>(
        accum, bc, sl, (size_t)N * HID);
    xin = sl;
    K = HID;
  }

  // ================= Bi-LSTM over L=3 =================
  const float* attW = (const float*)d_in[22];   // [1, 192]
  const float* attB = (const float*)d_in[23];
  for (int dir = 0; dir < 2; ++dir) {
    const float* Wih = (const float*)d_in[14 + 4 * dir];  // [384, 64]
    const float* Whh = (const float*)d_in[15 + 4 * dir];  // [384, 96]
    const float* bih = (const float*)d_in[16 + 4 * dir];
    const float* bhh = (const float*)d_in[17 + 4 * dir];
    float* lout = dir ? lb : lf;
    const float* aw = attW + dir * LSTMH;
    fill(lh, 0.f, (size_t)N * LSTMH);
    fill(lc, 0.f, (size_t)N * LSTMH);
    for (int s = 0; s < 3; ++s) {
      int t = dir ? (2 - s) : s;   // backward scan runs reversed in time
      gemm(seq + (size_t)t * N * HID, Wih, bih, bhh, gates, N, HID, G4, 0, 0);
      gemm(lh, Whh, nullptr, nullptr, gates, N, LSTMH, G4, 1, 0);  // +=
      lstm_pw_k<<<(unsigned)(((size_t)N * LSTMH + 255) / 256), 256, 0, stream>>>(
          gates, lh, lc, N);
      logit_k<<<(unsigned)(((size_t)N * 32 + 255) / 256), 256, 0, stream>>>(
          lh, aw, lout + (size_t)t * N, N);
    }
  }

  // ================= JK attention + output projection =================
  attn_k<<<(unsigned)(((size_t)N * 32 + 255) / 256), 256, 0, stream>>>(
      lf, lb, attB, seq, accum, N);
  gemm(accum, (const float*)d_in[24], (const float*)d_in[25], nullptr,
       (float*)d_out, N, HID, HID, 0, 0);
}